// stackedIndRNN_encoder_64123861729616
// MI455X (gfx1250) — compile-verified
//
#include <hip/hip_runtime.h>

// ---------------------------------------------------------------------------
// Stacked IndRNN encoder for MI455X (gfx1250), fp32 WMMA path.
// T=128, B=64, IN*ND=144, H=2048, L=4, C=10
// GEMM: block tile 128x64, 8 wave32, wave tile 32x32 (4 WMMA accumulators)
// -> 4 v_wmma_f32_16x16x4_f32 per 4 v2f fragment loads (1.0 wmma per ds_load).
// ---------------------------------------------------------------------------

typedef __attribute__((ext_vector_type(2))) float v2f;
typedef __attribute__((ext_vector_type(4))) float f32x4;
typedef __attribute__((ext_vector_type(8))) float v8f;

#define T_    128
#define B_    64
#define IN_   144
#define H_    2048
#define L_    4
#define C_    10
#define MROWS (T_ * B_)           // 8192 rows per GEMM
#define BNEPS 1e-5f

#define BLK_M 128
#define BLK_N 64
#define KT    32
#define LDT   36                  // padded LDS row stride (floats): 16B aligned, bank-spread

// ---------------------------------------------------------------------------
// GEMM:  Z[m,n] = sum_k A[m,k] * W[n,k] + bias[n]
// A: [M,K] row-major activations, W: [N,K] row-major weights.
// ---------------------------------------------------------------------------
__global__ __launch_bounds__(256) void gemm_bias_wmma(
    const float* __restrict__ A, const float* __restrict__ W,
    const float* __restrict__ bias, float* __restrict__ Z,
    int M, int N, int K)
{
    __shared__ float As[BLK_M * LDT];   // 18 KB
    __shared__ float Bs[BLK_N * LDT];   //  9 KB

    const int bm   = blockIdx.x * BLK_M;
    const int bn   = blockIdx.y * BLK_N;
    const int tid  = threadIdx.x;
    const int wave = tid >> 5;          // wave32: 8 waves / block
    const int lane = tid & 31;
    const int mi   = wave >> 1;         // 0..3 -> 32-row M sub-block
    const int ni   = wave & 1;          // 0..1 -> 32-col N sub-block
    const int ln   = lane & 15;
    const int kh   = (lane >> 4) << 1;  // lanes 16..31 hold K+2,K+3 (ISA A/B layout)

    v8f acc00 = {0.f,0.f,0.f,0.f,0.f,0.f,0.f,0.f};
    v8f acc01 = {0.f,0.f,0.f,0.f,0.f,0.f,0.f,0.f};
    v8f acc10 = {0.f,0.f,0.f,0.f,0.f,0.f,0.f,0.f};
    v8f acc11 = {0.f,0.f,0.f,0.f,0.f,0.f,0.f,0.f};

    for (int k0 = 0; k0 < K; k0 += KT) {
        // ---- Stage A (128x32) and B (64x32) tiles into LDS -----------------
        // A: 1024 float4 chunks (4/thread); B: 512 float4 chunks (2/thread).
#pragma unroll
        for (int i = 0; i < 4; ++i) {
            int idx = tid + (i << 8);         // 0..1023
            int row = idx >> 3;               // 0..127
            int c4  = (idx & 7) << 2;         // 0,4,...,28
            f32x4 va = {0.f,0.f,0.f,0.f};
            if (k0 + c4 < K)                  // K is a multiple of 4 (144 / 2048)
                va = *(const f32x4*)(A + (size_t)(bm + row) * K + k0 + c4);
            *(f32x4*)(As + row * LDT + c4) = va;
        }
#pragma unroll
        for (int i = 0; i < 2; ++i) {
            int idx = tid + (i << 8);         // 0..511
            int row = idx >> 3;               // 0..63
            int c4  = (idx & 7) << 2;
            f32x4 vb = {0.f,0.f,0.f,0.f};
            if (k0 + c4 < K)
                vb = *(const f32x4*)(W + (size_t)(bn + row) * K + k0 + c4);
            *(f32x4*)(Bs + row * LDT + c4) = vb;
        }
        // Prefetch next K-tile into cache (global_prefetch_b8 on gfx1250).
        if (k0 + KT < K) {
            __builtin_prefetch(A + (size_t)(bm + (tid >> 1)) * K + k0 + KT +
                                   ((tid & 1) << 4), 0, 1);
            __builtin_prefetch(W + (size_t)(bn + (tid >> 2)) * K + k0 + KT +
                                   ((tid & 3) << 3), 0, 1);
        }
        __syncthreads();

        const float* ap0 = As + (mi * 32 + ln) * LDT + kh;
        const float* ap1 = As + (mi * 32 + 16 + ln) * LDT + kh;
        const float* bp0 = Bs + (ni * 32 + ln) * LDT + kh;
        const float* bp1 = Bs + (ni * 32 + 16 + ln) * LDT + kh;
#pragma unroll
        for (int kk = 0; kk < KT; kk += 4) {
            v2f a0; a0.x = ap0[kk]; a0.y = ap0[kk + 1];
            v2f a1; a1.x = ap1[kk]; a1.y = ap1[kk + 1];
            v2f b0; b0.x = bp0[kk]; b0.y = bp0[kk + 1];
            v2f b1; b1.x = bp1[kk]; b1.y = bp1[kk + 1];
            acc00 = __builtin_amdgcn_wmma_f32_16x16x4_f32(
                false, a0, false, b0, (short)0, acc00, false, false);
            acc01 = __builtin_amdgcn_wmma_f32_16x16x4_f32(
                false, a0, false, b1, (short)0, acc01, false, false);
            acc10 = __builtin_amdgcn_wmma_f32_16x16x4_f32(
                false, a1, false, b0, (short)0, acc10, false, false);
            acc11 = __builtin_amdgcn_wmma_f32_16x16x4_f32(
                false, a1, false, b1, (short)0, acc11, false, false);
        }
        __syncthreads();
    }

    // C/D layout: VGPR r -> M = r + 8*(lane>=16), N = lane%16.
    const int hi   = (lane >> 4) << 3;        // 0 or 8
    const int r0   = bm + mi * 32 + hi;       // M-sub 0 rows
    const int r1   = r0 + 16;                 // M-sub 1 rows
    const int c0   = bn + ni * 32 + ln;       // N-sub 0 col
    const int c1   = c0 + 16;                 // N-sub 1 col
    const float bv0 = bias[c0];
    const float bv1 = bias[c1];
#pragma unroll
    for (int r = 0; r < 8; ++r) {
        Z[(size_t)(r0 + r) * N + c0] = acc00[r] + bv0;
        Z[(size_t)(r0 + r) * N + c1] = acc01[r] + bv1;
        Z[(size_t)(r1 + r) * N + c0] = acc10[r] + bv0;
        Z[(size_t)(r1 + r) * N + c1] = acc11[r] + bv1;
    }
}

// ---------------------------------------------------------------------------
// IndRNN diagonal recurrence, in place on Z ([T,B,H] row-major):
//   h_t = relu(z_t + u*h_{t-1});  also accumulates per-channel sum / sumsq.
// One thread per (b, channel) -> fully coalesced along H.
// ---------------------------------------------------------------------------
__global__ __launch_bounds__(256) void indrnn_scan(
    float* __restrict__ Z, const float* __restrict__ u,
    float* __restrict__ sums, float* __restrict__ sumsq)
{
    const int idx = blockIdx.x * 256 + threadIdx.x;   // 0 .. B*H-1
    const int ch  = idx & (H_ - 1);
    const float uu = u[ch];
    float h = 0.f, s = 0.f, s2 = 0.f;
    float* p = Z + idx;
#pragma unroll 4
    for (int t = 0; t < T_; ++t) {
        float z = p[(size_t)t * (B_ * H_)];
        h = fmaxf(fmaf(uu, h, z), 0.f);
        p[(size_t)t * (B_ * H_)] = h;
        s += h;
        s2 = fmaf(h, h, s2);
    }
    atomicAdd(sums + ch, s);
    atomicAdd(sumsq + ch, s2);
}

__global__ __launch_bounds__(256) void zero_stats(float* __restrict__ p, int n)
{
    int i = blockIdx.x * 256 + threadIdx.x;
    if (i < n) p[i] = 0.f;
}

// ---------------------------------------------------------------------------
// Training-mode BatchNorm over (T,B) per channel, biased variance (jnp.var).
// ---------------------------------------------------------------------------
__global__ __launch_bounds__(256) void bn_apply(
    float* __restrict__ Hbuf, const float* __restrict__ sums,
    const float* __restrict__ sumsq, const float* __restrict__ gamma,
    const float* __restrict__ beta)
{
    const size_t i = (size_t)blockIdx.x * 256 + threadIdx.x;
    const int ch = (int)(i & (H_ - 1));
    const float inv = 1.0f / (float)(T_ * B_);
    const float m = sums[ch] * inv;
    const float v = sumsq[ch] * inv - m * m;
    Hbuf[i] = (Hbuf[i] - m) * rsqrtf(v + BNEPS) * gamma[ch] + beta[ch];
}

// ---------------------------------------------------------------------------
// Head: out[b,c] = sum_h hlast[b,h] * Wlast[c,h] + blast[c]   (64x10, tiny)
// ---------------------------------------------------------------------------
__global__ __launch_bounds__(64) void head_kernel(
    const float* __restrict__ Hlast, const float* __restrict__ Wl,
    const float* __restrict__ bl, float* __restrict__ out)
{
    const int b = threadIdx.x;
    float acc[C_];
#pragma unroll
    for (int c = 0; c < C_; ++c) acc[c] = 0.f;
    const float* hp = Hlast + (size_t)b * H_;
    for (int h = 0; h < H_; ++h) {
        float x = hp[h];
#pragma unroll
        for (int c = 0; c < C_; ++c)
            acc[c] = fmaf(x, Wl[(size_t)c * H_ + h], acc[c]);
    }
#pragma unroll
    for (int c = 0; c < C_; ++c) out[b * C_ + c] = acc[c] + bl[c];
}

// ---------------------------------------------------------------------------
// Orchestration. Workspace: buf0 (64MB) | buf1 (64MB) | sums (8KB) | sumsq (8KB)
// ---------------------------------------------------------------------------
extern "C" void kernel_launch(void* const* d_in, const int* in_sizes, int n_in,
                              void* d_out, int out_size, void* d_ws, size_t ws_size,
                              hipStream_t stream)
{
    (void)in_sizes; (void)n_in; (void)out_size; (void)ws_size;

    const float* x      = (const float*)d_in[0];   // [T,B,144]
    const float* W0     = (const float*)d_in[1];   // [H,144]
    const float* Ws     = (const float*)d_in[2];   // [L-1,H,H]
    const float* bs     = (const float*)d_in[3];   // [L,H]
    const float* us     = (const float*)d_in[4];   // [L,H]
    const float* gammas = (const float*)d_in[5];   // [L,H]
    const float* betas  = (const float*)d_in[6];   // [L,H]
    const float* Wlast  = (const float*)d_in[7];   // [C,H]
    const float* blast  = (const float*)d_in[8];   // [C]
    float* out = (float*)d_out;

    const size_t actN = (size_t)T_ * B_ * H_;      // 16,777,216 floats
    float* buf0  = (float*)d_ws;
    float* buf1  = buf0 + actN;
    float* sums  = buf1 + actN;
    float* sumsq = sums + H_;

    const dim3 gg(MROWS / BLK_M, H_ / BLK_N);      // (64, 32)

    float* cur = nullptr;
    for (int l = 0; l < L_; ++l) {
        const float* Ain = (l == 0) ? x : cur;
        const float* Wl  = (l == 0) ? W0 : (Ws + (size_t)(l - 1) * H_ * H_);
        const int    K   = (l == 0) ? IN_ : H_;
        float* zbuf = (l & 1) ? buf1 : buf0;

        gemm_bias_wmma<<<gg, 256, 0, stream>>>(Ain, Wl, bs + (size_t)l * H_,
                                               zbuf, MROWS, H_, K);
        zero_stats<<<(2 * H_ + 255) / 256, 256, 0, stream>>>(sums, 2 * H_);
        indrnn_scan<<<(B_ * H_) / 256, 256, 0, stream>>>(zbuf, us + (size_t)l * H_,
                                                         sums, sumsq);
        bn_apply<<<(unsigned)(actN / 256), 256, 0, stream>>>(
            zbuf, sums, sumsq, gammas + (size_t)l * H_, betas + (size_t)l * H_);
        cur = zbuf;
    }

    head_kernel<<<1, 64, 0, stream>>>(cur + (size_t)(T_ - 1) * B_ * H_,
                                      Wlast, blast, out);
}